// MFAMBlock_47227460387000
// MI455X (gfx1250) — compile-verified
//
#include <hip/hip_runtime.h>
#include <math.h>

typedef float v2f __attribute__((ext_vector_type(2)));
typedef float v8f __attribute__((ext_vector_type(8)));

#define NC 3  // channel count

// ---------------------------------------------------------------------------
// Kernel 1: per-pixel 1x1 conv (C=3 channel matmul) + bias + PReLU using
// V_WMMA_F32_16X16X4_F32. One wave32 handles 16 pixels.
//   A (16x4 f32): M = pixel, K = in-channel (3 real + 1 zero pad)
//     ISA layout: lanes 0-15 hold M=lane {VGPR0=K0, VGPR1=K1},
//                 lanes 16-31 hold M=lane-16 {VGPR0=K2, VGPR1=K3}
//   B (4x16 f32): K = in-channel, N = out-channel (3 real, rest zero)
//     row-striped across lanes: VGPR0 lanes0-15=K0row, VGPR1 lanes0-15=K1row,
//                               VGPR0 lanes16-31=K2row, VGPR1 lanes16-31=K3row
//   D (16x16 f32): VGPR r -> M = r (lanes 0-15) / r+8 (lanes 16-31), N = lane&15
// All weight/bias/alpha reads are wave-uniform (scalar loads), selected per
// lane with cndmask — no divergent loads feeding the WMMA.
// stride=2 implements the ::2,::2 subsample of down_sample; stride=1 for up.
// ---------------------------------------------------------------------------
__global__ __launch_bounds__(256) void conv1x1_prelu_wmma(
    const float* __restrict__ in, float* __restrict__ out,
    const float* __restrict__ Wg, const float* __restrict__ bias,
    const float* __restrict__ alpha, int Hout, int Wout, int stride)
{
  const int lane = threadIdx.x & 31;
  const int wave = threadIdx.x >> 5;
  const int n16  = lane & 15;                 // M index for A, N index for D
  const bool hi  = lane >= 16;
  const int pixelBase = (blockIdx.x * 8 + wave) * 16;

  // ---- uniform parameter loads (scalar path) ----
  const float w00 = Wg[0], w01 = Wg[1], w02 = Wg[2];
  const float w10 = Wg[3], w11 = Wg[4], w12 = Wg[5];
  const float w20 = Wg[6], w21 = Wg[7], w22 = Wg[8];
  const float b0 = bias[0], b1 = bias[1], b2 = bias[2];
  const float al0 = alpha[0], al1 = alpha[1], al2 = alpha[2];

  // ---- A: activations of pixel pixelBase+n16 ----
  const int pix = pixelBase + n16;
  const int w  = pix % Wout;
  const int t  = pix / Wout;
  const int h  = t % Hout;
  const int b  = t / Hout;
  const int inH = Hout * stride, inW = Wout * stride;
  const float* ip = in + ((b * inH + h * stride) * inW + w * stride) * NC;
  v2f A;
  A.x = hi ? ip[2] : ip[0];
  A.y = hi ? 0.0f  : ip[1];

  // ---- B: select column n16 of each padded weight row (uniform data) ----
  const float wn0 = (n16 == 0) ? w00 : (n16 == 1) ? w01 : (n16 == 2) ? w02 : 0.0f;
  const float wn1 = (n16 == 0) ? w10 : (n16 == 1) ? w11 : (n16 == 2) ? w12 : 0.0f;
  const float wn2 = (n16 == 0) ? w20 : (n16 == 1) ? w21 : (n16 == 2) ? w22 : 0.0f;
  v2f Bm;
  Bm.x = hi ? wn2 : wn0;
  Bm.y = hi ? 0.0f : wn1;

  v8f Cm = {};
  // (neg_a, A, neg_b, B, c_mod, C, reuse_a, reuse_b)
  v8f D = __builtin_amdgcn_wmma_f32_16x16x4_f32(false, A, false, Bm,
                                                (short)0, Cm, false, false);

  // ---- bias + PReLU + scatter store (only the 3 valid N columns) ----
  const float bn = (n16 == 0) ? b0 : (n16 == 1) ? b1 : b2;
  const float an = (n16 == 0) ? al0 : (n16 == 1) ? al1 : al2;
  if (n16 < NC) {
    const int mOff = hi ? 8 : 0;
#pragma unroll
    for (int r = 0; r < 8; ++r) {
      float y = D[r] + bn;
      y = (y >= 0.0f) ? y : an * y;
      out[(pixelBase + mOff + r) * NC + n16] = y;
    }
  }
}

// ---------------------------------------------------------------------------
// s(b,p,q,c) = maxpool2x2s1(a1) + input2 + max(prelu(b_up), upsampled e)
// cb must be prelu(b_up[c], alpha_up[c]).
// ---------------------------------------------------------------------------
__device__ __forceinline__ float compute_s(
    const float* __restrict__ a1, const float* __restrict__ e,
    const float* __restrict__ in2, float cb,
    int b, int p, int q, int c, int H, int He)
{
  const float* a1b = a1 + b * H * H * NC;
  float i1 = a1b[(p * H + q) * NC + c];
  if (q + 1 < H)   i1 = fmaxf(i1, a1b[(p * H + q + 1) * NC + c]);
  if (p + 1 < H) {
    i1 = fmaxf(i1, a1b[((p + 1) * H + q) * NC + c]);
    if (q + 1 < H) i1 = fmaxf(i1, a1b[((p + 1) * H + q + 1) * NC + c]);
  }
  const int hp = (p + 1) >> 1, wq = (q + 1) >> 1;
  float i3 = cb;                       // every 2x2 window holds a const-bias cell
  if (hp < He && wq < He)
    i3 = fmaxf(i3, e[((b * He + hp) * He + wq) * NC + c]);
  const float i2 = in2[((b * H + p) * H + q) * NC + c];
  return i1 + i2 + i3;
}

// ---------------------------------------------------------------------------
// Kernel 2: compute s on the fly, per-block partial sums per channel
// (deterministic: fixed-order LDS tree, no float atomics).
// One block = 256 consecutive pixels of a single batch (H*H % 256 == 0).
// ---------------------------------------------------------------------------
__global__ __launch_bounds__(256) void s_partial_sums(
    const float* __restrict__ a1, const float* __restrict__ e,
    const float* __restrict__ in2, const float* __restrict__ b_up,
    const float* __restrict__ al_up, float* __restrict__ part,
    int H, int He)
{
  __shared__ float red[NC * 256];
  const int tid = threadIdx.x;
  const int pix = blockIdx.x * 256 + tid;
  const int q = pix % H;
  const int t = pix / H;
  const int p = t % H;
  const int b = t / H;
#pragma unroll
  for (int c = 0; c < NC; ++c) {
    const float bu = b_up[c];
    const float cb = (bu >= 0.0f) ? bu : al_up[c] * bu;
    red[c * 256 + tid] = compute_s(a1, e, in2, cb, b, p, q, c, H, He);
  }
  __syncthreads();
  for (int off = 128; off > 0; off >>= 1) {
    if (tid < off) {
#pragma unroll
      for (int c = 0; c < NC; ++c)
        red[c * 256 + tid] += red[c * 256 + tid + off];
    }
    __syncthreads();
  }
  if (tid == 0) {
#pragma unroll
    for (int c = 0; c < NC; ++c) part[blockIdx.x * NC + c] = red[c * 256];
  }
}

// ---------------------------------------------------------------------------
// Kernel 3: one block per batch. Deterministic reduction of partial sums,
// then mean -> conv1x1+PReLU -> conv1x1 -> softmax gate y (3 values/batch).
// ---------------------------------------------------------------------------
__global__ __launch_bounds__(256) void gate_kernel(
    const float* __restrict__ part, const float* __restrict__ W_c,
    const float* __restrict__ b_c, const float* __restrict__ alpha_g,
    float* __restrict__ yv, int blocksPerBatch, float invHW)
{
  __shared__ float red[NC * 256];
  const int tid = threadIdx.x;
  const int b = blockIdx.x;
  float acc[NC] = {0.0f, 0.0f, 0.0f};
  for (int i = tid; i < blocksPerBatch; i += 256) {
    const float* pp = part + (b * blocksPerBatch + i) * NC;
#pragma unroll
    for (int c = 0; c < NC; ++c) acc[c] += pp[c];
  }
#pragma unroll
  for (int c = 0; c < NC; ++c) red[c * 256 + tid] = acc[c];
  __syncthreads();
  for (int off = 128; off > 0; off >>= 1) {
    if (tid < off) {
#pragma unroll
      for (int c = 0; c < NC; ++c)
        red[c * 256 + tid] += red[c * 256 + tid + off];
    }
    __syncthreads();
  }
  if (tid == 0) {
    float g[NC], x[NC], z[NC];
#pragma unroll
    for (int c = 0; c < NC; ++c) g[c] = red[c * 256] * invHW;
#pragma unroll
    for (int d = 0; d < NC; ++d) {
      float v = b_c[d];
#pragma unroll
      for (int c = 0; c < NC; ++c) v += g[c] * W_c[c * NC + d];
      x[d] = (v >= 0.0f) ? v : alpha_g[d] * v;
    }
#pragma unroll
    for (int d = 0; d < NC; ++d) {
      float v = b_c[d];
#pragma unroll
      for (int c = 0; c < NC; ++c) v += x[c] * W_c[c * NC + d];
      z[d] = v;
    }
    const float mx = fmaxf(z[0], fmaxf(z[1], z[2]));
    const float e0 = expf(z[0] - mx), e1 = expf(z[1] - mx), e2 = expf(z[2] - mx);
    const float inv = 1.0f / (e0 + e1 + e2);
    yv[b * NC + 0] = e0 * inv;
    yv[b * NC + 1] = e1 * inv;
    yv[b * NC + 2] = e2 * inv;
  }
}

// ---------------------------------------------------------------------------
// Kernel 4: out = y[b,c] * s(b,p,q,c); s recomputed (a1/e hot in 192MB L2).
// ---------------------------------------------------------------------------
__global__ __launch_bounds__(256) void scale_out_kernel(
    const float* __restrict__ a1, const float* __restrict__ e,
    const float* __restrict__ in2, const float* __restrict__ b_up,
    const float* __restrict__ al_up, const float* __restrict__ yv,
    float* __restrict__ out, int H, int He, int Ntot)
{
  const int i = blockIdx.x * 256 + threadIdx.x;
  if (i >= Ntot) return;
  const int c = i % NC;
  const int pix = i / NC;
  const int q = pix % H;
  const int t = pix / H;
  const int p = t % H;
  const int b = t / H;
  const float bu = b_up[c];
  const float cb = (bu >= 0.0f) ? bu : al_up[c] * bu;
  const float s = compute_s(a1, e, in2, cb, b, p, q, c, H, He);
  out[i] = yv[b * NC + c] * s;
}

extern "C" void kernel_launch(void* const* d_in, const int* in_sizes, int n_in,
                              void* d_out, int out_size, void* d_ws, size_t ws_size,
                              hipStream_t stream) {
  (void)n_in; (void)out_size; (void)ws_size;
  const float* input1  = (const float*)d_in[0];
  const float* input2  = (const float*)d_in[1];
  const float* input3  = (const float*)d_in[2];
  const float* W_down  = (const float*)d_in[3];
  const float* b_down  = (const float*)d_in[4];
  const float* a_down  = (const float*)d_in[5];
  const float* W_up    = (const float*)d_in[6];
  const float* b_up    = (const float*)d_in[7];
  const float* a_up    = (const float*)d_in[8];
  const float* W_c     = (const float*)d_in[9];
  const float* b_c     = (const float*)d_in[10];
  const float* alpha_g = (const float*)d_in[11];
  // d_in[12] = scale (==2), path hardcoded per reference assert.

  const int H  = 512, He = 256;
  const int B  = in_sizes[1] / (H * H * NC);   // 8
  const int Npix  = B * H * H;                 // 2,097,152 (multiple of 128)
  const int NpixE = B * He * He;               //   524,288 (multiple of 128)

  // workspace layout (floats): a1 | e | per-block partials | y gate
  float* a1   = (float*)d_ws;
  float* e    = a1 + (size_t)Npix * NC;
  float* part = e  + (size_t)NpixE * NC;
  const int nBlk2 = Npix / 256;                // 8192 blocks, 1024 per batch
  float* yv   = part + (size_t)nBlk2 * NC;

  // 1) a1 = prelu(subsample(input1) @ W_down + b_down)   [WMMA]
  conv1x1_prelu_wmma<<<Npix / 128, 256, 0, stream>>>(
      input1, a1, W_down, b_down, a_down, H, H, /*stride=*/2);
  // 2) e = prelu(input3 @ W_up + b_up)                   [WMMA]
  conv1x1_prelu_wmma<<<NpixE / 128, 256, 0, stream>>>(
      input3, e, W_up, b_up, a_up, He, He, /*stride=*/1);
  // 3) partial sums of s per (block, channel)
  s_partial_sums<<<nBlk2, 256, 0, stream>>>(a1, e, input2, b_up, a_up, part, H, He);
  // 4) gate y per (batch, channel)
  gate_kernel<<<B, 256, 0, stream>>>(part, W_c, b_c, alpha_g, yv,
                                     nBlk2 / B, 1.0f / (float)(H * H));
  // 5) out = y * s
  const int Ntot = Npix * NC;
  scale_out_kernel<<<(Ntot + 255) / 256, 256, 0, stream>>>(
      a1, e, input2, b_up, a_up, yv, (float*)d_out, H, He, Ntot);
}